// GraphConvolution_23218593202729
// MI455X (gfx1250) — compile-verified
//
#include <hip/hip_runtime.h>
#include <hip/hip_bf16.h>

typedef __attribute__((ext_vector_type(2))) float v2f;
typedef __attribute__((ext_vector_type(8))) float v8f;

#define F_IN  256
#define F_OUT 256

// ---------------------------------------------------------------------------
// Kernel 1: out[n, f] = b[f]   (bias pre-fill; segment_sum accumulates on top)
// One thread writes one float4. n_nodes * 64 threads total.
// ---------------------------------------------------------------------------
__global__ __launch_bounds__(256) void gcn_bias_init_kernel(
    const float* __restrict__ bias, float* __restrict__ out, int n_nodes) {
    long long i = (long long)blockIdx.x * blockDim.x + threadIdx.x;
    long long total = (long long)n_nodes * (F_OUT / 4);
    if (i >= total) return;
    int fg = (int)(i & (F_OUT / 4 - 1));          // feature group 0..63
    float4 bb = *(const float4*)(bias + fg * 4);
    *(float4*)(out + i * 4) = bb;                  // contiguous layout matches
}

// ---------------------------------------------------------------------------
// Kernel 2: support = X @ W using V_WMMA_F32_16X16X4_F32 (fp32 matrix pipe).
// Block = 256 threads (8 waves). Block tile = 32 rows x 256 cols.
//   wave w: rows base+16*(w&1), cols (w>>1)*64  -> 4 accumulator tiles (16x64).
// K loop steps by 4 (WMMA K), 64 iterations; A fragment reused across 4 WMMAs.
// ---------------------------------------------------------------------------
__global__ __launch_bounds__(256) void gcn_gemm_wmma_kernel(
    const float* __restrict__ X, const float* __restrict__ W,
    float* __restrict__ S, int n_nodes) {
    const int lane = threadIdx.x & 31;
    const int wave = threadIdx.x >> 5;
    const int row0 = blockIdx.x * 32 + (wave & 1) * 16;
    const int col0 = (wave >> 1) * 64;
    if (row0 >= n_nodes) return;

    const int m  = lane & 15;   // row within A tile / col within B tile
    const int kk = lane >> 4;   // K half-select (0 or 1)

    v8f acc0 = {}, acc1 = {}, acc2 = {}, acc3 = {};

    const float* __restrict__ arow = X + (size_t)(row0 + m) * F_IN;

    for (int k = 0; k < F_IN; k += 4) {
        const int ka = k + 2 * kk;  // this lane-half's K pair: ka, ka+1
        // A fragment: lane holds A[m][ka], A[m][ka+1]  (one 64-bit load)
        v2f a = *(const v2f*)(arow + ka);
        // B fragments: lane holds W[ka][n], W[ka+1][n] for each 16-col tile
        const float* __restrict__ wr0 = W + (size_t)ka * F_OUT + col0 + m;
        const float* __restrict__ wr1 = wr0 + F_OUT;
        v2f b0; b0[0] = wr0[0];  b0[1] = wr1[0];
        v2f b1; b1[0] = wr0[16]; b1[1] = wr1[16];
        v2f b2; b2[0] = wr0[32]; b2[1] = wr1[32];
        v2f b3; b3[0] = wr0[48]; b3[1] = wr1[48];

        acc0 = __builtin_amdgcn_wmma_f32_16x16x4_f32(
            false, a, false, b0, (short)0, acc0, false, false);
        acc1 = __builtin_amdgcn_wmma_f32_16x16x4_f32(
            false, a, false, b1, (short)0, acc1, false, false);
        acc2 = __builtin_amdgcn_wmma_f32_16x16x4_f32(
            false, a, false, b2, (short)0, acc2, false, false);
        acc3 = __builtin_amdgcn_wmma_f32_16x16x4_f32(
            false, a, false, b3, (short)0, acc3, false, false);
    }

    // C/D layout: VGPR r -> M = r + 8*(lane>>4), N = lane&15
    const int crow = row0 + (lane >> 4) * 8;
    const int ccol = col0 + (lane & 15);
#pragma unroll
    for (int r = 0; r < 8; ++r) {
        float* __restrict__ orow = S + (size_t)(crow + r) * F_OUT + ccol;
        orow[0]  = acc0[r];
        orow[16] = acc1[r];
        orow[32] = acc2[r];
        orow[48] = acc3[r];
    }
}

// ---------------------------------------------------------------------------
// Kernel 3: edge scatter.  out[rows[e], :] += vals[e] * support[cols[e], :]
// 64 lanes per edge, each lane handles 4 features (b128 gather + 4 atomics).
// support (102 MB) and out (102 MB) both fit the 192 MB L2, so gathers and
// float atomics resolve on-chip.
// ---------------------------------------------------------------------------
__global__ __launch_bounds__(256) void gcn_scatter_kernel(
    const long long* __restrict__ rows, const long long* __restrict__ cols,
    const float* __restrict__ vals, const float* __restrict__ S,
    float* __restrict__ out, int n_edges) {
    const int t    = threadIdx.x & 63;           // feature group (4 floats)
    const int esub = threadIdx.x >> 6;           // 4 edges per block
    const int e    = blockIdx.x * 4 + esub;
    if (e >= n_edges) return;

    const long long r = rows[e];
    const long long c = cols[e];
    const float     v = vals[e];

    const float4 s = *(const float4*)(S + c * F_OUT + t * 4);
    float* __restrict__ o = out + r * F_OUT + t * 4;
    atomicAdd(o + 0, v * s.x);
    atomicAdd(o + 1, v * s.y);
    atomicAdd(o + 2, v * s.z);
    atomicAdd(o + 3, v * s.w);
}

// ---------------------------------------------------------------------------
extern "C" void kernel_launch(void* const* d_in, const int* in_sizes, int n_in,
                              void* d_out, int out_size, void* d_ws, size_t ws_size,
                              hipStream_t stream) {
    const float*      x     = (const float*)d_in[0];
    const long long*  erows = (const long long*)d_in[1];   // int64 in reference
    const long long*  ecols = (const long long*)d_in[2];   // int64 in reference
    const float*      evals = (const float*)d_in[3];
    const float*      w     = (const float*)d_in[4];
    const float*      b     = (const float*)d_in[5];
    float*            out   = (float*)d_out;
    float*            supp  = (float*)d_ws;                 // n_nodes*F_OUT floats

    const int n_nodes = in_sizes[0] / F_IN;
    const int n_edges = in_sizes[1];

    // 1) out = broadcast(bias)
    {
        long long total = (long long)n_nodes * (F_OUT / 4);
        int grid = (int)((total + 255) / 256);
        gcn_bias_init_kernel<<<grid, 256, 0, stream>>>(b, out, n_nodes);
    }
    // 2) support = X @ W  (fp32 WMMA)
    {
        int grid = (n_nodes + 31) / 32;
        gcn_gemm_wmma_kernel<<<grid, 256, 0, stream>>>(x, w, supp, n_nodes);
    }
    // 3) out += scatter-add over edges
    {
        int grid = (n_edges + 3) / 4;
        gcn_scatter_kernel<<<grid, 256, 0, stream>>>(erows, ecols, evals, supp,
                                                     out, n_edges);
    }
}